// FullyConnectedAttention_82806969466907
// MI455X (gfx1250) — compile-verified
//
#include <hip/hip_runtime.h>
#include <hip/hip_bf16.h>
#include <math.h>

typedef __attribute__((ext_vector_type(2))) float v2f;
typedef __attribute__((ext_vector_type(8))) float v8f;

#define NROWS 8192
#define DIM   512

// ---------------------------------------------------------------------------
// K1: H = X (8192x512) * W (512x512), fp32, via V_WMMA_F32_16X16X4_F32.
// Block = 128 threads (4 waves). Each wave computes one 16x16 tile of H.
// blockIdx.x -> M tile (512 tiles), blockIdx.y -> group of 4 N tiles (8 groups).
// A layout (16x4 f32): lanes 0-15 hold M=0..15 with K={k,k+1} in the 2 VGPRs,
// lanes 16-31 hold K={k+2,k+3}.  B layout (4x16 f32): lane%16 = N, VGPR r holds
// row K = k + (lane/16)*2 + r.  C/D: lane%16 = N, VGPR r = row r + (lane/16)*8.
// ---------------------------------------------------------------------------
__global__ void __launch_bounds__(128)
hgemm_wmma(const float* __restrict__ X, const float* __restrict__ W,
           float* __restrict__ H) {
  const int wave = threadIdx.x >> 5;
  const int lane = threadIdx.x & 31;
  const int half = lane >> 4;   // 0: lanes 0-15, 1: lanes 16-31
  const int lm   = lane & 15;   // M index for A, N index for B/C/D

  const int m0 = blockIdx.x << 4;
  const int n0 = (blockIdx.y << 6) + (wave << 4);

  const float* __restrict__ xrow = X + (size_t)(m0 + lm) * DIM;
  const float* __restrict__ wcol = W + (n0 + lm);

  v8f acc = {0.f, 0.f, 0.f, 0.f, 0.f, 0.f, 0.f, 0.f};

#pragma unroll 4
  for (int k = 0; k < DIM; k += 4) {
    const int ka = k + (half << 1);
    const v2f a = *(const v2f*)(xrow + ka);   // A[m][ka], A[m][ka+1] (8B aligned)
    v2f b;
    b.x = wcol[ka * DIM];                     // B[ka][n]
    b.y = wcol[(ka + 1) * DIM];               // B[ka+1][n]
    acc = __builtin_amdgcn_wmma_f32_16x16x4_f32(
        /*neg_a=*/false, a, /*neg_b=*/false, b,
        /*c_mod=*/(short)0, acc, /*reuse_a=*/false, /*reuse_b=*/false);
  }

  float* __restrict__ hout = H + n0 + lm;
#pragma unroll
  for (int r = 0; r < 8; ++r) {
    hout[(size_t)(m0 + r + (half << 3)) * DIM] = acc[r];
  }
}

// ---------------------------------------------------------------------------
// K2: s2[row] = dot(H[row,:], a2)   (a2 = attention_vector[512:1024])
// One wave per row; H is L2-resident (16 MB << 192 MB).
// ---------------------------------------------------------------------------
__global__ void __launch_bounds__(256)
s2_kernel(const float* __restrict__ H, const float* __restrict__ av,
          float* __restrict__ s2) {
  const int row  = blockIdx.x * 8 + (threadIdx.x >> 5);
  const int lane = threadIdx.x & 31;
  const float* __restrict__ hr = H + (size_t)row * DIM;
  const float* __restrict__ a2 = av + DIM;
  float acc = 0.f;
#pragma unroll
  for (int c = lane; c < DIM; c += 32) acc = fmaf(hr[c], a2[c], acc);
#pragma unroll
  for (int off = 16; off; off >>= 1) acc += __shfl_xor(acc, off);
  if (lane == 0) s2[row] = acc;
}

// ---------------------------------------------------------------------------
// K3: stats[0] = max(s2), stats[1] = 1/sum(exp(s2-max)); also zero out_row
// (workspace is poisoned with 0xAA, and K4 accumulates with atomics).
// ---------------------------------------------------------------------------
__global__ void __launch_bounds__(1024)
softmax_stats_kernel(const float* __restrict__ s2, float* __restrict__ stats,
                     float* __restrict__ out_row) {
  __shared__ float red[32];
  __shared__ float bmax;
  const int t = threadIdx.x;

  float m = -1e30f;
  for (int i = t; i < NROWS; i += 1024) m = fmaxf(m, s2[i]);
#pragma unroll
  for (int off = 16; off; off >>= 1) m = fmaxf(m, __shfl_xor(m, off));
  if ((t & 31) == 0) red[t >> 5] = m;
  __syncthreads();
  if (t < 32) {
    float v = red[t];
#pragma unroll
    for (int off = 16; off; off >>= 1) v = fmaxf(v, __shfl_xor(v, off));
    if (t == 0) bmax = v;
  }
  __syncthreads();
  const float mx = bmax;

  float s = 0.f;
  for (int i = t; i < NROWS; i += 1024) s += expf(s2[i] - mx);
#pragma unroll
  for (int off = 16; off; off >>= 1) s += __shfl_xor(s, off);
  if ((t & 31) == 0) red[t >> 5] = s;
  __syncthreads();
  if (t < 32) {
    float v = red[t];
#pragma unroll
    for (int off = 16; off; off >>= 1) v += __shfl_xor(v, off);
    if (t == 0) { stats[0] = mx; stats[1] = 1.0f / v; }
  }
  if (t < DIM) out_row[t] = 0.0f;
}

// ---------------------------------------------------------------------------
// K4: out_row[c] = sum_j p[j] * H[j][c],  p[j] = exp(s2[j]-max)/sum.
// 64 blocks x 128 rows each; each thread owns columns c and c+256.
// ---------------------------------------------------------------------------
__global__ void __launch_bounds__(256)
colsum_kernel(const float* __restrict__ H, const float* __restrict__ s2,
              const float* __restrict__ stats, float* __restrict__ out_row) {
  const int c  = threadIdx.x;
  const int r0 = blockIdx.x * 128;
  const float mx  = stats[0];
  const float inv = stats[1];
  float acc0 = 0.f, acc1 = 0.f;
  for (int j = 0; j < 128; ++j) {
    const int row = r0 + j;
    const float p = expf(s2[row] - mx) * inv;
    const float* __restrict__ hr = H + (size_t)row * DIM;
    acc0 = fmaf(p, hr[c], acc0);
    acc1 = fmaf(p, hr[c + 256], acc1);
  }
  atomicAdd(out_row + c, acc0);
  atomicAdd(out_row + c + 256, acc1);
}

// ---------------------------------------------------------------------------
// K5: out[i][:] = out_row[:] for all i (16 MB of float4 stores, coalesced).
// ---------------------------------------------------------------------------
__global__ void __launch_bounds__(256)
bcast_kernel(const float* __restrict__ out_row, float4* __restrict__ out) {
  const int idx = blockIdx.x * 256 + threadIdx.x;   // 8192*512/4 = 1M float4s
  const int c4  = idx & 127;                        // column group within a row
  out[idx] = ((const float4*)out_row)[c4];
}

extern "C" void kernel_launch(void* const* d_in, const int* in_sizes, int n_in,
                              void* d_out, int out_size, void* d_ws, size_t ws_size,
                              hipStream_t stream) {
  const float* X  = (const float*)d_in[0];   // node_features  (8192 x 512)
  const float* W  = (const float*)d_in[1];   // weight_matrix  (512 x 512)
  const float* av = (const float*)d_in[2];   // attention_vector (1024 x 1)
  float* out = (float*)d_out;

  char* ws = (char*)d_ws;
  float* H       = (float*)ws;                                   // 16 MB
  float* s2      = (float*)(ws + (size_t)NROWS * DIM * sizeof(float));
  float* stats   = s2 + NROWS;                                   // 2 floats (pad 16)
  float* out_row = stats + 16;                                   // 512 floats

  dim3 g1(NROWS / 16, DIM / 64);
  hgemm_wmma<<<g1, 128, 0, stream>>>(X, W, H);
  s2_kernel<<<NROWS / 8, 256, 0, stream>>>(H, av, s2);
  softmax_stats_kernel<<<1, 1024, 0, stream>>>(s2, stats, out_row);
  colsum_kernel<<<NROWS / 128, 256, 0, stream>>>(H, s2, stats, out_row);
  bcast_kernel<<<(NROWS * DIM / 4) / 256, 256, 0, stream>>>(out_row, (float4*)out);
}